// GCN_28621662060799
// MI455X (gfx1250) — compile-verified
//
#include <hip/hip_runtime.h>

// GCN forward on MI455X (gfx1250), wave32.
//   support = X @ W              -> fp32 WMMA 16x16x4 (full precision)
//   out     = SpMM(A,support)+b  -> wave-per-row segmented reduction (deterministic)
//
// Roofline: SpMM gathers dominate (2 x 3.2M x 256B ~ 1.6GB, served from the
// 192MB L2 since the 25.6MB table is resident); HBM traffic ~180MB total.
// GEMM (1.64 GFLOP) is microseconds on the WMMA path; W is staged in LDS with
// a k-pair-interleaved layout so each B fragment is one aligned ds_load_b64
// landing directly in the WMMA operand register pair (no assembly movs).

typedef __attribute__((ext_vector_type(2))) float v2f;
typedef __attribute__((ext_vector_type(8))) float v8f;

#define DIM 64

// ---------------------------------------------------------------------------
// Dense GEMM  Y[n x 64] = X[n x 64] * W[64 x 64]  via V_WMMA_F32_16X16X4_F32.
// Block = 256 threads = 8 waves; each wave owns a 16-row x 64-col output tile.
// LDS layout: Ws2[kp*64 + c] = (W[2kp][c], W[2kp+1][c])  (float2, 16KB).
// A frag (16x4 f32): lanes 0-15 = rows M; VGPR0/1 = K{0,1} (lanes<16) and
// K{2,3} (lanes>=16). B frag mirrors with N on lanes. C/D: 8 VGPRs, VGPR r =
// row r (lanes<16) / row r+8 (lanes>=16), col = lane&15.
// ---------------------------------------------------------------------------
__global__ __launch_bounds__(256) void gemm_wmma_f32(
    const float* __restrict__ X, const float* __restrict__ W,
    float* __restrict__ Y, int nrows)
{
    __shared__ v2f Ws2[(DIM / 2) * DIM];   // 2048 pairs = 16KB
    for (int p = threadIdx.x; p < (DIM / 2) * DIM; p += 256) {
        const int kp = p >> 6;             // k-pair index 0..31
        const int c  = p & 63;             // column 0..63
        v2f w;
        w.x = W[(2 * kp    ) * DIM + c];   // coalesced across c
        w.y = W[(2 * kp + 1) * DIM + c];
        Ws2[p] = w;                        // ds_store_b64
    }
    __syncthreads();

    const int wave = threadIdx.x >> 5;
    const int lane = threadIdx.x & 31;
    const int half = lane >> 4;            // 0: K{0,1}, 1: K{2,3}
    const int l16  = lane & 15;
    const int row_base = blockIdx.x * 128 + wave * 16;
    if (row_base >= nrows) return;         // after the only barrier: safe

    const float* __restrict__ xrow = X + (size_t)(row_base + l16) * DIM;

    v8f acc0 = {}, acc1 = {}, acc2 = {}, acc3 = {};
#pragma unroll
    for (int s = 0; s < 16; ++s) {
        const int k0 = s * 4 + half * 2;   // even
        const int kp = k0 >> 1;            // = 2s + half
        const v2f a = *(const v2f*)(xrow + k0);          // global_load_b64
        const v2f* __restrict__ brow = Ws2 + kp * DIM + l16;
        const v2f b0 = brow[0];                          // ds_load_b64
        const v2f b1 = brow[16];
        const v2f b2 = brow[32];
        const v2f b3 = brow[48];
        acc0 = __builtin_amdgcn_wmma_f32_16x16x4_f32(false, a, false, b0, (short)0, acc0, false, false);
        acc1 = __builtin_amdgcn_wmma_f32_16x16x4_f32(false, a, false, b1, (short)0, acc1, false, false);
        acc2 = __builtin_amdgcn_wmma_f32_16x16x4_f32(false, a, false, b2, (short)0, acc2, false, false);
        acc3 = __builtin_amdgcn_wmma_f32_16x16x4_f32(false, a, false, b3, (short)0, acc3, false, false);
    }

    float* __restrict__ ybase = Y + (size_t)row_base * DIM;
#pragma unroll
    for (int r = 0; r < 8; ++r) {
        const int m = r + half * 8;
        if (row_base + m < nrows) {
            float* yr = ybase + (size_t)m * DIM + l16;
            yr[0]  = acc0[r];
            yr[16] = acc1[r];
            yr[32] = acc2[r];
            yr[48] = acc3[r];
        }
    }
}

// ---------------------------------------------------------------------------
// SpMM: out[i] = sum_{e: row[e]==i} val[e] * S[col[e]]  (+bias, opt LeakyReLU)
// row is sorted -> each wave binary-searches its row's edge range, then the
// 32 lanes cooperatively load 32 edges at a time and broadcast via __shfl.
// Each lane keeps 2 output features (lane, lane+32). Deterministic order.
// ---------------------------------------------------------------------------
__device__ inline int lower_bound_i64(const long long* __restrict__ a, int n,
                                      long long key)
{
    int lo = 0, hi = n;
    while (lo < hi) {
        int mid = (lo + hi) >> 1;
        if (a[mid] < key) lo = mid + 1; else hi = mid;
    }
    return lo;
}

template <bool LEAKY>
__global__ __launch_bounds__(256) void spmm_row_wave(
    const long long* __restrict__ row, const long long* __restrict__ col,
    const float* __restrict__ val, const float* __restrict__ S,
    const float* __restrict__ bias, float* __restrict__ O,
    int n_nodes, int n_edges)
{
    const int wave = threadIdx.x >> 5;
    const int lane = threadIdx.x & 31;
    const int i = blockIdx.x * 8 + wave;
    if (i >= n_nodes) return;

    const int start = lower_bound_i64(row, n_edges, (long long)i);
    const int end   = lower_bound_i64(row, n_edges, (long long)i + 1);

    float acc0 = 0.f, acc1 = 0.f;
    for (int e0 = start; e0 < end; e0 += 32) {
        if (e0 + 64 <= n_edges) {                 // prefetch next edge chunk
            __builtin_prefetch(&col[e0 + 32], 0, 1);   // global_prefetch_b8
            __builtin_prefetch(&val[e0 + 32], 0, 1);
        }
        const int e = e0 + lane;
        int   c = 0;
        float v = 0.f;
        if (e < end) {
            c = (int)col[e];
            v = val[e];
        }
        const int m = min(32, end - e0);
        for (int t = 0; t < m; ++t) {
            const int   ct = __shfl(c, t, 32);
            const float vt = __shfl(v, t, 32);
            const float* __restrict__ sr = S + (size_t)ct * DIM;
            acc0 = fmaf(vt, sr[lane],      acc0);   // coalesced 128B
            acc1 = fmaf(vt, sr[lane + 32], acc1);   // coalesced 128B
        }
    }

    float o0 = acc0 + bias[lane];
    float o1 = acc1 + bias[lane + 32];
    if (LEAKY) {
        o0 = (o0 >= 0.f) ? o0 : 0.2f * o0;
        o1 = (o1 >= 0.f) ? o1 : 0.2f * o1;
    }
    float* __restrict__ orow = O + (size_t)i * DIM;
    orow[lane]      = o0;
    orow[lane + 32] = o1;
}

// ---------------------------------------------------------------------------
extern "C" void kernel_launch(void* const* d_in, const int* in_sizes, int n_in,
                              void* d_out, int out_size, void* d_ws, size_t ws_size,
                              hipStream_t stream)
{
    const float*     x    = (const float*)d_in[0];
    const long long* arow = (const long long*)d_in[1];
    const long long* acol = (const long long*)d_in[2];
    const float*     aval = (const float*)d_in[3];
    const float*     W1   = (const float*)d_in[4];
    const float*     b1   = (const float*)d_in[5];
    const float*     W2   = (const float*)d_in[6];
    const float*     b2   = (const float*)d_in[7];
    float*           out  = (float*)d_out;

    const int n_nodes = in_sizes[0] / DIM;
    const int n_edges = in_sizes[1];

    float* support = (float*)d_ws;                        // n_nodes * 64 floats
    float* h       = support + (size_t)n_nodes * DIM;     // n_nodes * 64 floats

    dim3 blk(256);
    dim3 ggrid((n_nodes + 127) / 128);   // 128 rows per block (8 waves x 16)
    dim3 sgrid((n_nodes + 7) / 8);       // 8 rows per block (1 wave per row)

    // layer 1: support = X@W1 ; h = leaky(SpMM(support)+b1)
    gemm_wmma_f32<<<ggrid, blk, 0, stream>>>(x, W1, support, n_nodes);
    spmm_row_wave<true><<<sgrid, blk, 0, stream>>>(arow, acol, aval, support, b1,
                                                   h, n_nodes, n_edges);
    // layer 2: support = h@W2 ; out = SpMM(support)+b2
    gemm_wmma_f32<<<ggrid, blk, 0, stream>>>(h, W2, support, n_nodes);
    spmm_row_wave<false><<<sgrid, blk, 0, stream>>>(arow, acol, aval, support, b2,
                                                    out, n_nodes, n_edges);
}